// BackProjectionLoss_64604898066751
// MI455X (gfx1250) — compile-verified
//
#include <hip/hip_runtime.h>

// ---------------------------------------------------------------------------
// BackProjectionLoss for MI455X (gfx1250, wave32)
//
// bicubic downsample output(32,3,512,512) -> (32,3,128,128) (16-tap, stride 4,
// symmetric pad 6, separable), then mean(|down - input|).
//
// Workgroup = 4 waves = 4 horizontally adjacent 16x16 output tiles.
// Stage 0: async-copy the shared 80x272 (reflected) source region into LDS,
//          TRANSPOSED (col-major) so WMMA B K-pairs are single ds_load_b64.
// Stage 1: vertical conv  V(16x80) = Aw(16x80) x X  via V_WMMA_F32_16X16X4_F32
// Stage 2: horizontal conv D(16x16) = V x Wh        via V_WMMA_F32_16X16X4_F32
// Stage 3: |D - input|, wave reduce, atomicAdd of pre-scaled partials.
// ---------------------------------------------------------------------------

typedef __attribute__((ext_vector_type(2))) float v2f;
typedef __attribute__((ext_vector_type(8))) float v8f;

#if defined(__has_builtin)
#if __has_builtin(__builtin_amdgcn_global_load_async_to_lds_b32)
#define HAVE_ASYNC_LDS 1
#endif
#endif

typedef __attribute__((address_space(1))) int* gptr_t;  // global (device) int*
typedef __attribute__((address_space(3))) int* lptr_t;  // LDS int*

// cubic weight; taps 0..15 give |arg| <= 1.875 so only two pieces needed
constexpr double cub(double ax) {
  return (ax <= 1.0) ? ((1.5 * ax - 2.5) * ax * ax + 1.0)
                     : (((-0.5) * ax + 2.5) * ax * ax - 4.0 * ax + 2.0);
}
constexpr double norm_calc() {
  double s = 0.0;
  for (int k = 0; k < 16; ++k) {
    double d = 7.5 - (double)k;
    double ax = (d < 0 ? -d : d) * 0.25;
    s += cub(ax);
  }
  return 1.0 / s;
}
constexpr float NORMF = (float)norm_calc();

__device__ __forceinline__ float wcoef(int d) {  // w[d], d in [0,16), else 0
  if (d < 0 || d > 15) return 0.0f;
  float ax = fabsf(7.5f - (float)d) * 0.25f;
  float w = (ax <= 1.0f) ? ((1.5f * ax - 2.5f) * ax * ax + 1.0f)
                         : (((-0.5f) * ax + 2.5f) * ax * ax - 4.0f * ax + 2.0f);
  return w * NORMF;
}

__device__ __forceinline__ int refl(int v) {  // symmetric reflection, len 512
  v = (v < 0) ? (-v - 1) : v;
  v = (v > 511) ? (1023 - v) : v;
  return v;
}

#define XS 82                    // Xt row stride (floats): even -> b64-aligned
#define VS 84                    // V row stride (floats)
#define XT_FLOATS (272 * XS)     // staged region: 272 cols x 80 rows (transposed)
#define SMEM_BYTES ((XT_FLOATS + 4 * 16 * VS) * 4)

__global__ __launch_bounds__(128) void bploss_kernel(
    const float* __restrict__ ref,   // (32,3,128,128)
    const float* __restrict__ img,   // (32,3,512,512)
    float* __restrict__ out)
{
  extern __shared__ float smem[];
  float* Xt  = smem;                       // Xt[c*XS + r], c in [0,272), r in [0,80)
  float* lsV = smem + XT_FLOATS;           // 4 waves x (16 x VS)

  const int tid  = threadIdx.x;
  const int lane = tid & 31;
  const int wid  = tid >> 5;               // 0..3 -> tile within group

  const int b  = blockIdx.x;               // 0..1535
  const int bc = b >> 4;                   // 0..95
  const int ti = (b >> 1) & 7;             // row tile
  const int tg = b & 1;                    // column tile group (tj = 4*tg + wid)

  const int m  = lane & 15;
  const int hi = lane >> 4;

  const float* imgp = img + (size_t)bc * (512 * 512);
  const int rowbase = ti * 64 - 6;         // virtual source row base
  const int colb0   = tg * 256 - 6;        // virtual source col base of group

  // ---- Stage 0: stage 80x272 reflected region into LDS, transposed --------
  // flat index i over 80*272 = 21760 elements; 128 threads -> 170 each, no tail
  {
    int r = 0, c = tid;                    // i = r*272 + c
    for (int it = 0; it < 170; ++it) {
      const int srow = refl(rowbase + r);
      const int scol = refl(colb0 + c);
      const float* gp = imgp + (size_t)srow * 512 + scol;
      float* lp = &Xt[c * XS + r];
#if HAVE_ASYNC_LDS
      __builtin_amdgcn_global_load_async_to_lds_b32((gptr_t)gp, (lptr_t)lp, 0, 0);
#else
      *lp = *gp;
#endif
      c += 128;
      if (c >= 272) { c -= 272; ++r; }
    }
  }
#if HAVE_ASYNC_LDS
#if __has_builtin(__builtin_amdgcn_s_wait_asynccnt)
  __builtin_amdgcn_s_wait_asynccnt(0);
#else
  asm volatile("s_wait_asynccnt 0x0" ::: "memory");
#endif
#endif
  __syncthreads();

  // ---- Stage 1: vertical conv via WMMA ------------------------------------
  // V[r,q] = sum_K Aw[r,K] * X[K, q],  Aw[r,K] = w[K-4r]
  float* Vt = lsV + wid * (16 * VS);
  const int xcb = wid * 64;                // this tile's col offset in Xt
  for (int cs = 0; cs < 5; ++cs) {
    const int cb = cs * 16;
    const float* xcol = &Xt[(xcb + cb + m) * XS];   // per-lane source column
    v8f acc = {};
#pragma unroll
    for (int t = 0; t < 20; ++t) {
      const int K0 = 4 * t + 2 * hi;       // VGPR0 K; VGPR1 holds K0+1
      v2f a;
      a.x = wcoef(K0     - 4 * m);         // A[m, K0]
      a.y = wcoef(K0 + 1 - 4 * m);
      v2f bfrag = *(const v2f*)&xcol[K0];  // X[K0, col], X[K0+1, col] (b64)
      acc = __builtin_amdgcn_wmma_f32_16x16x4_f32(
          false, a, false, bfrag, (short)0, acc, false, false);
    }
#pragma unroll
    for (int p = 0; p < 8; ++p)            // D: VGPR p -> M=p+8*hi, N=m
      Vt[(p + 8 * hi) * VS + cb + m] = acc[p];
  }
  // V produced and consumed by the same wave: DScnt ordering only.

  // ---- Stage 2: horizontal conv via WMMA ----------------------------------
  // D[r,j] = sum_q V[r,q] * Wh[q,j],  Wh[q,j] = w[q-4j]
  v8f acc2 = {};
#pragma unroll
  for (int t = 0; t < 20; ++t) {
    const int q0 = 4 * t + 2 * hi;
    v2f a = *(const v2f*)&Vt[m * VS + q0]; // A[m,q0], A[m,q0+1] (b64)
    v2f bfrag;
    bfrag.x = wcoef(q0     - 4 * m);       // B[q0,   m]
    bfrag.y = wcoef(q0 + 1 - 4 * m);
    acc2 = __builtin_amdgcn_wmma_f32_16x16x4_f32(
        false, a, false, bfrag, (short)0, acc2, false, false);
  }

  // ---- Stage 3: loss -------------------------------------------------------
  const int tj = tg * 4 + wid;
  const float* refp = ref + (size_t)bc * (128 * 128)
                          + (size_t)(ti * 16) * 128 + tj * 16;
  float s = 0.0f;
#pragma unroll
  for (int p = 0; p < 8; ++p)
    s += fabsf(acc2[p] - refp[(p + 8 * hi) * 128 + m]);
  s *= (1.0f / 1572864.0f);                // 32*3*128*128

#pragma unroll
  for (int o = 16; o > 0; o >>= 1) s += __shfl_down(s, o, 32);
  if (lane == 0) atomicAdd(out, s);
}

extern "C" void kernel_launch(void* const* d_in, const int* in_sizes, int n_in,
                              void* d_out, int out_size, void* d_ws, size_t ws_size,
                              hipStream_t stream) {
  (void)in_sizes; (void)n_in; (void)out_size; (void)d_ws; (void)ws_size;
  const float* ref = (const float*)d_in[0];   // 'input'  (32,3,128,128)
  const float* img = (const float*)d_in[1];   // 'output' (32,3,512,512)
  float* out = (float*)d_out;

  (void)hipMemsetAsync(out, 0, sizeof(float), stream);
  // 96 bc x 8 ti x 2 tile-groups = 1536 blocks of 128 threads (4 waves/tiles)
  bploss_kernel<<<1536, 128, SMEM_BYTES, stream>>>(ref, img, out);
}